// GCN_24008867184689
// MI455X (gfx1250) — compile-verified
//
#include <hip/hip_runtime.h>

#define N_NODES 50000
#define N_EDGES 800000
#define FEATS   128

typedef __attribute__((ext_vector_type(2))) float v2f;
typedef __attribute__((ext_vector_type(4))) float v4f;
typedef __attribute__((ext_vector_type(8))) float v8f;

// ---------------------------------------------------------------------------
// Kernel 1: zero msg_sum[N_NODES*FEATS] and deg[N_NODES] (ws is poisoned 0xAA)
// ---------------------------------------------------------------------------
__global__ __launch_bounds__(256) void gcn_zero(float* __restrict__ msg,
                                                float* __restrict__ deg) {
    int i = blockIdx.x * 256 + threadIdx.x;
    if (i < N_NODES * FEATS) msg[i] = 0.0f;
    if (i < N_NODES)         deg[i] = 0.0f;
}

// ---------------------------------------------------------------------------
// Kernel 2: edge scatter. One wave (32 lanes) per edge; each lane moves a
// float4 (4 lanes * 32 = 128 feats) via hardware f32 atomics (no-return ->
// global_atomic_add_f32, STOREcnt). Lane 0 bumps the in-degree.
// ---------------------------------------------------------------------------
__global__ __launch_bounds__(256) void gcn_scatter(const float* __restrict__ feat,
                                                   const int*   __restrict__ src,
                                                   const int*   __restrict__ dst,
                                                   float* __restrict__ msg,
                                                   float* __restrict__ deg) {
    int lane = threadIdx.x & 31;
    int wave = threadIdx.x >> 5;
    long long e = (long long)blockIdx.x * 8 + wave;
    if (e >= N_EDGES) return;

    int s = src[e];
    int d = dst[e];

    v4f v = *(const v4f*)(feat + (long long)s * FEATS + lane * 4);
    float* p = msg + (long long)d * FEATS + lane * 4;
    (void)__hip_atomic_fetch_add(p + 0, v.x, __ATOMIC_RELAXED, __HIP_MEMORY_SCOPE_AGENT);
    (void)__hip_atomic_fetch_add(p + 1, v.y, __ATOMIC_RELAXED, __HIP_MEMORY_SCOPE_AGENT);
    (void)__hip_atomic_fetch_add(p + 2, v.z, __ATOMIC_RELAXED, __HIP_MEMORY_SCOPE_AGENT);
    (void)__hip_atomic_fetch_add(p + 3, v.w, __ATOMIC_RELAXED, __HIP_MEMORY_SCOPE_AGENT);
    if (lane == 0)
        (void)__hip_atomic_fetch_add(deg + d, 1.0f, __ATOMIC_RELAXED, __HIP_MEMORY_SCOPE_AGENT);
}

// ---------------------------------------------------------------------------
// Kernel 3: h = deg>0 ? msg/deg : feature   (in-place into msg buffer)
// ---------------------------------------------------------------------------
__global__ __launch_bounds__(256) void gcn_mean(const float* __restrict__ feat,
                                                const float* __restrict__ deg,
                                                float* __restrict__ msg) {
    int i = blockIdx.x * 256 + threadIdx.x;
    if (i >= N_NODES * FEATS) return;
    int n = i >> 7;                       // FEATS == 128
    float dg = deg[n];
    msg[i] = (dg > 0.0f) ? (msg[i] / dg) : feat[i];
}

// ---------------------------------------------------------------------------
// Kernel 4: out = relu(h @ W + b) via V_WMMA_F32_16X16X4_F32 (full fp32).
// One wave computes one 16x16 output tile; 8 waves/block cover the 8 N-tiles
// of a 16-row stripe. K=128 -> 32 chained WMMAs.
//
// Fragment layouts (ISA 7.12.2, 32-bit, wave32):
//   A 16x4 : lanes 0-15 -> M=lane,    VGPR0=K+0, VGPR1=K+1
//            lanes16-31 -> M=lane-16, VGPR0=K+2, VGPR1=K+3
//   B 4x16 : lanes 0-15 -> N=lane,    VGPR0=K+0, VGPR1=K+1
//            lanes16-31 -> N=lane-16, VGPR0=K+2, VGPR1=K+3
//   C/D    : VGPR r, lanes 0-15 -> (M=r,   N=lane)
//                    lanes16-31 -> (M=r+8, N=lane-16)
// ---------------------------------------------------------------------------
__global__ __launch_bounds__(256) void gcn_gemm_wmma(const float* __restrict__ h,
                                                     const float* __restrict__ W,
                                                     const float* __restrict__ bias,
                                                     float* __restrict__ out) {
    int lane  = threadIdx.x & 31;
    int wave  = threadIdx.x >> 5;   // 0..7 -> N tile
    int m0    = blockIdx.x * 16;
    int n0    = wave * 16;
    int row   = lane & 15;
    int hi    = lane >> 4;          // 0 or 1
    int khalf = hi * 2;             // K offset 0 or 2 within the 4-wide slab

    float bv = bias[n0 + row];
    v8f c = {bv, bv, bv, bv, bv, bv, bv, bv};

    const float* arow = h + (long long)(m0 + row) * FEATS;

#pragma unroll
    for (int k = 0; k < FEATS; k += 4) {
        v2f a = *(const v2f*)(arow + k + khalf);                 // A[M=row][K=k+khalf .. +1]
        v2f bb;
        bb.x = W[(k + khalf)     * FEATS + n0 + row];            // B[K=k+khalf  ][N]
        bb.y = W[(k + khalf + 1) * FEATS + n0 + row];            // B[K=k+khalf+1][N]
        c = __builtin_amdgcn_wmma_f32_16x16x4_f32(
                /*neg_a=*/false, a, /*neg_b=*/false, bb,
                /*c_mod=*/(short)0, c, /*reuse_a=*/false, /*reuse_b=*/false);
    }

    long long obase = (long long)(m0 + hi * 8) * FEATS + n0 + row;
#pragma unroll
    for (int r = 0; r < 8; ++r)
        out[obase + (long long)r * FEATS] = fmaxf(c[r], 0.0f);
}

// ---------------------------------------------------------------------------
extern "C" void kernel_launch(void* const* d_in, const int* in_sizes, int n_in,
                              void* d_out, int out_size, void* d_ws, size_t ws_size,
                              hipStream_t stream) {
    (void)in_sizes; (void)n_in; (void)out_size; (void)ws_size;

    const float* feat = (const float*)d_in[0];
    const int*   src  = (const int*)  d_in[1];
    const int*   dst  = (const int*)  d_in[2];
    const float* W    = (const float*)d_in[3];
    const float* b    = (const float*)d_in[4];
    float*       out  = (float*)d_out;

    float* msg = (float*)d_ws;                        // N_NODES*FEATS f32 (reused as h)
    float* deg = msg + (size_t)N_NODES * FEATS;       // N_NODES f32

    const int total  = N_NODES * FEATS;
    const int zgrid  = (total + 255) / 256;

    gcn_zero   <<<zgrid, 256, 0, stream>>>(msg, deg);
    gcn_scatter<<<(N_EDGES + 7) / 8, 256, 0, stream>>>(feat, src, dst, msg, deg);
    gcn_mean   <<<zgrid, 256, 0, stream>>>(feat, deg, msg);
    gcn_gemm_wmma<<<N_NODES / 16, 256, 0, stream>>>(msg, W, b, out);
}